// SolverInTheLoop_13022340841764
// MI455X (gfx1250) — compile-verified
//
#include <hip/hip_runtime.h>

// ---------------- Physics constants (mirroring the reference) ----------------
constexpr double PI_D    = 3.14159265358979323846;
constexpr float  INV_H   = (float)(1.0 / 0.3);
constexpr float  MASS    = (float)(0.3 * 0.3 * 0.3);            // DX^3
constexpr float  INV_MASS= (float)(1.0 / (0.3 * 0.3 * 0.3));
constexpr float  SIGMA   = (float)(3.0 / (359.0 * PI_D * 0.3 * 0.3 * 0.3));
constexpr float  SIGMA_H = (float)((3.0 / (359.0 * PI_D * 0.3 * 0.3 * 0.3)) / 0.3);
constexpr float  P_REF   = 100.0f;                               // rho_ref*c^2/gamma
constexpr float  ETA_IJ  = (float)(2.0 * 0.01 * 0.01 / (0.01 + 0.01 + 1e-8)); // const visc
constexpr float  INV_DT  = 1000.0f;                              // u = vel_hist / dt

// ---------------- TDM staging geometry ----------------
constexpr int CHUNK  = 1024;        // edges per TDM tile (4 KB per index array)
constexpr int NCHBLK = 8;           // chunks per block
constexpr int EPB    = CHUNK * NCHBLK; // 8192 edges per block
constexpr int TPB    = 256;         // 8 waves (wave32)

typedef __attribute__((ext_vector_type(4))) unsigned int v4u;
typedef __attribute__((ext_vector_type(8))) int          v8i;
typedef __attribute__((ext_vector_type(4))) int          v4i;

// Build a D# for a 1-D contiguous copy of `tile_elems` dwords from global to
// LDS byte offset `lds_off`. `valid_elems` bounds the tensor so tail reads
// return zero. Layout per CDNA5 ISA.
__device__ __forceinline__ void tdm_load_ints(const int* gptr, unsigned lds_off,
                                              unsigned tile_elems, unsigned valid_elems) {
  unsigned long long ga = (unsigned long long)(const void*)gptr;
  v4u g0;
  g0[0] = 1u;                                                   // count=1, user mode
  g0[1] = lds_off;                                              // lds_addr (bytes)
  g0[2] = (unsigned)ga;                                         // global_addr[31:0]
  g0[3] = (unsigned)((ga >> 32) & 0x01FFFFFFu) | (2u << 30);    // addr[56:32] | type=2
  v8i g1;
  g1[0] = (int)(2u << 16);                                      // wg_mask=0, data_size=4B
  g1[1] = (int)((valid_elems & 0xFFFFu) << 16);                 // tensor_dim0[15:0]
  g1[2] = (int)((valid_elems >> 16) & 0xFFFFu) | (1 << 16);     // dim0[31:16] | tensor_dim1=1
  g1[3] = (int)((tile_elems & 0xFFFFu) << 16);                  // tile_dim0
  g1[4] = 0;                                                    // tile_dim1=0, tile_dim2=0
  g1[5] = (int)tile_elems;                                      // tensor_dim0_stride (unused)
  g1[6] = 0;
  g1[7] = 0;
  v4i z4 = {0, 0, 0, 0};
#if defined(__clang_major__) && (__clang_major__ >= 23)
  v8i z8 = {0, 0, 0, 0, 0, 0, 0, 0};
  __builtin_amdgcn_tensor_load_to_lds(g0, g1, z4, z4, z8, 0);
#else
  __builtin_amdgcn_tensor_load_to_lds(g0, g1, z4, z4, 0);
#endif
}

// The TDM engine writes LDS behind the compiler's back; clang sees a shared
// array with no stores and folds its loads to undef. Escaping the pointer
// through an opaque asm with a memory clobber forces the compiler to treat
// the buffer as externally written.
__device__ __forceinline__ void lds_mark_written(void* p) {
  asm volatile("" : "+v"(p) : : "memory");
}

// Group-relative LDS byte offset of a __shared__ object: the generic (flat)
// address of an LDS variable is aperture_base | group_offset; hardware uses
// addr[31:0] as the LDS offset.
__device__ __forceinline__ unsigned lds_offset_of(const void* p) {
  return (unsigned)(unsigned long long)p;
}

struct F3 { float x, y, z; };
__device__ __forceinline__ F3 ld3(const float* __restrict__ base, int idx) {
  const float* p = base + 3 * idx;
  F3 v; v.x = p[0]; v.y = p[1]; v.z = p[2]; return v;
}

__device__ __forceinline__ float quintic_w(float d) {
  float q  = d * INV_H;
  float q1 = fmaxf(0.f, 1.f - q), q2 = fmaxf(0.f, 2.f - q), q3 = fmaxf(0.f, 3.f - q);
  float a = q1 * q1, b = q2 * q2, c = q3 * q3;
  float q15 = a * a * q1, q25 = b * b * q2, q35 = c * c * q3;
  return SIGMA * (q35 - 6.f * q25 + 15.f * q15);
}
__device__ __forceinline__ float quintic_gw(float d) {
  float q  = d * INV_H;
  float q1 = fmaxf(0.f, 1.f - q), q2 = fmaxf(0.f, 2.f - q), q3 = fmaxf(0.f, 3.f - q);
  float a = q1 * q1, b = q2 * q2, c = q3 * q3;
  return SIGMA_H * (-5.f * c * c + 30.f * b * b - 75.f * a * a);
}

// ---------------- Kernels ----------------
__global__ void __launch_bounds__(TPB) sitl_zero(float* __restrict__ out, int n) {
  int i = blockIdx.x * blockDim.x + threadIdx.x;
  if (i < n) out[i] = 0.0f;
}

__global__ void __launch_bounds__(TPB) sitl_density(const float* __restrict__ r,
                                                    const int* __restrict__ i_s,
                                                    const int* __restrict__ j_s,
                                                    float* __restrict__ rho, int E) {
  __shared__ int sh[4 * CHUNK];   // [buf][i|j][CHUNK]
  const unsigned sh_base = lds_offset_of(sh);
  const int base = blockIdx.x * EPB;
  if (base >= E) return;
  const int nEdges  = min(EPB, E - base);
  const int nChunks = (nEdges + CHUNK - 1) / CHUNK;

  auto issue = [&](int c) {
    const int cs = base + c * CHUNK;
    const unsigned v = (unsigned)min(CHUNK, E - cs);
    const int b = c & 1;
    tdm_load_ints(i_s + cs, sh_base + (unsigned)((2 * b) * CHUNK * 4), CHUNK, v);
    tdm_load_ints(j_s + cs, sh_base + (unsigned)((2 * b + 1) * CHUNK * 4), CHUNK, v);
  };

  if (threadIdx.x == 0) { issue(0); __builtin_amdgcn_s_wait_tensorcnt(0); }
  __syncthreads();

  for (int c = 0; c < nChunks; ++c) {
    if (threadIdx.x == 0 && c + 1 < nChunks) issue(c + 1);
    lds_mark_written(sh);                              // TDM wrote this buffer
    const int b = c & 1;
    const int v = min(CHUNK, nEdges - c * CHUNK);
    for (int t = (int)threadIdx.x; t < v; t += TPB) {
      int i = sh[(2 * b) * CHUNK + t];
      int j = sh[(2 * b + 1) * CHUNK + t];
      F3 ri = ld3(r, i), rj = ld3(r, j);
      float dx = ri.x - rj.x, dy = ri.y - rj.y, dz = ri.z - rj.z;
      float d = sqrtf(dx * dx + dy * dy + dz * dz);
      unsafeAtomicAdd(&rho[i], MASS * quintic_w(d));
    }
    __syncthreads();                                   // readers done with both bufs
    if (threadIdx.x == 0 && c + 1 < nChunks) __builtin_amdgcn_s_wait_tensorcnt(0);
    __syncthreads();                                   // next chunk visible in LDS
  }
}

__global__ void __launch_bounds__(TPB) sitl_eos(const float* __restrict__ rho,
                                                float* __restrict__ p, int n) {
  int i = blockIdx.x * blockDim.x + threadIdx.x;
  if (i < n) p[i] = P_REF * (rho[i] - 1.0f);
}

__global__ void __launch_bounds__(TPB) sitl_accel(const float* __restrict__ r,
                                                  const float* __restrict__ vel,
                                                  const int* __restrict__ i_s,
                                                  const int* __restrict__ j_s,
                                                  const float* __restrict__ rho,
                                                  float* __restrict__ dudt, int E) {
  __shared__ int sh[4 * CHUNK];
  const unsigned sh_base = lds_offset_of(sh);
  const int base = blockIdx.x * EPB;
  if (base >= E) return;
  const int nEdges  = min(EPB, E - base);
  const int nChunks = (nEdges + CHUNK - 1) / CHUNK;

  auto issue = [&](int c) {
    const int cs = base + c * CHUNK;
    const unsigned v = (unsigned)min(CHUNK, E - cs);
    const int b = c & 1;
    tdm_load_ints(i_s + cs, sh_base + (unsigned)((2 * b) * CHUNK * 4), CHUNK, v);
    tdm_load_ints(j_s + cs, sh_base + (unsigned)((2 * b + 1) * CHUNK * 4), CHUNK, v);
  };

  if (threadIdx.x == 0) { issue(0); __builtin_amdgcn_s_wait_tensorcnt(0); }
  __syncthreads();

  for (int c = 0; c < nChunks; ++c) {
    if (threadIdx.x == 0 && c + 1 < nChunks) issue(c + 1);
    lds_mark_written(sh);
    const int b = c & 1;
    const int v = min(CHUNK, nEdges - c * CHUNK);
    for (int t = (int)threadIdx.x; t < v; t += TPB) {
      int i = sh[(2 * b) * CHUNK + t];
      int j = sh[(2 * b + 1) * CHUNK + t];
      F3 ri = ld3(r, i), rj = ld3(r, j);
      float dx = ri.x - rj.x, dy = ri.y - rj.y, dz = ri.z - rj.z;
      float d = sqrtf(dx * dx + dy * dy + dz * dz);
      float rho_i = rho[i], rho_j = rho[j];
      float p_i = P_REF * (rho_i - 1.0f);        // recompute EoS: 1 FLOP < 1 gather
      float p_j = P_REF * (rho_j - 1.0f);
      F3 ui = ld3(vel, i), uj = ld3(vel, j);
      float ux = (ui.x - uj.x) * INV_DT;
      float uy = (ui.y - uj.y) * INV_DT;
      float uz = (ui.z - uj.z) * INV_DT;
      float p_ij = (rho_j * p_i + rho_i * p_j) / (rho_i + rho_j);
      float mi = MASS / rho_i, mj = MASS / rho_j;
      float wvol = (mi * mi + mj * mj) * INV_MASS;
      float cc = wvol * quintic_gw(d) / (d + 1e-8f);
      float ax = cc * (-p_ij * dx + ETA_IJ * ux);
      float ay = cc * (-p_ij * dy + ETA_IJ * uy);
      float az = cc * (-p_ij * dz + ETA_IJ * uz);
      float* o = dudt + 3 * i;
      unsafeAtomicAdd(o + 0, ax);
      unsafeAtomicAdd(o + 1, ay);
      unsafeAtomicAdd(o + 2, az);
    }
    __syncthreads();
    if (threadIdx.x == 0 && c + 1 < nChunks) __builtin_amdgcn_s_wait_tensorcnt(0);
    __syncthreads();
  }
}

// ---------------- Host launcher ----------------
extern "C" void kernel_launch(void* const* d_in, const int* in_sizes, int n_in,
                              void* d_out, int out_size, void* d_ws, size_t ws_size,
                              hipStream_t stream) {
  const float* r   = (const float*)d_in[0];
  const float* vel = (const float*)d_in[1];
  const int*   i_s = (const int*)d_in[2];
  const int*   j_s = (const int*)d_in[3];
  const int N = in_sizes[0] / 3;
  const int E = in_sizes[2];

  float* out  = (float*)d_out;
  float* dudt = out;            // [3N]
  float* rho  = out + 3LL * N;  // [N]
  float* p    = out + 4LL * N;  // [N]

  const int zn = 4 * N;  // dudt + rho are atomic accumulators: zero every call
  sitl_zero<<<(zn + TPB - 1) / TPB, TPB, 0, stream>>>(out, zn);

  const int nblocks = (E + EPB - 1) / EPB;
  sitl_density<<<nblocks, TPB, 0, stream>>>(r, i_s, j_s, rho, E);
  sitl_eos<<<(N + TPB - 1) / TPB, TPB, 0, stream>>>(rho, p, N);
  sitl_accel<<<nblocks, TPB, 0, stream>>>(r, vel, i_s, j_s, rho, dudt, E);
}